// HyperScore_2310692405957
// MI455X (gfx1250) — compile-verified
//
#include <hip/hip_runtime.h>
#include <math.h>

#define B_ 32
#define J_ 1182
#define D_ 512
#define L_ 77
#define C_ 5

typedef __attribute__((ext_vector_type(2))) float v2f;
typedef __attribute__((ext_vector_type(8))) float v8f;

__device__ __forceinline__ float wave_red(float v) {
  #pragma unroll
  for (int off = 16; off > 0; off >>= 1) v += __shfl_xor(v, off, 32);
  return v;
}

__device__ __forceinline__ float sigmoidf(float x) { return 1.f / (1.f + expf(-x)); }

// ---------------------------------------------------------------- K1: condition normalize + gram loss
__global__ void k_cond(const float* __restrict__ cond, float* __restrict__ fn, float* __restrict__ out_loss) {
  __shared__ float invn[8];
  __shared__ __align__(16) float fnl[C_ * D_];
  __shared__ float acc;
  int tid = threadIdx.x, w = tid >> 5, lane = tid & 31;
  if (tid == 0) acc = 0.f;
  if (w < C_) {
    float ss = 0.f;
    const float4* rp = (const float4*)(cond + w * D_);
    #pragma unroll
    for (int g = 0; g < 4; ++g) {
      float4 v = rp[g * 32 + lane];
      ss += v.x * v.x + v.y * v.y + v.z * v.z + v.w * v.w;
    }
    ss = wave_red(ss);
    if (lane == 0) invn[w] = 1.f / fmaxf(sqrtf(ss), 1e-12f);
  }
  __syncthreads();
  for (int i = tid; i < C_ * D_; i += blockDim.x) {
    float v = cond[i] * invn[i >> 9];
    fn[i] = v; fnl[i] = v;
  }
  __syncthreads();
  const int pi[10] = {0,0,0,0,1,1,1,2,2,3};
  const int pj[10] = {1,2,3,4,2,3,4,3,4,4};
  if (w < 10) {
    float d = 0.f;
    for (int i = lane; i < D_; i += 32) d += fnl[pi[w] * D_ + i] * fnl[pj[w] * D_ + i];
    d = wave_red(d);
    if (lane == 0) atomicAdd(&acc, fmaxf(d, 0.f));
  }
  __syncthreads();
  if (tid == 0) out_loss[0] = acc * 0.1f;
}

// ---------------------------------------------------------------- K2: prompt row norms + sim_cond
__global__ void k_prompt(const float* __restrict__ prompt, const float* __restrict__ fn,
                         float* __restrict__ pinv, float* __restrict__ simc) {
  int tid = threadIdx.x, w = tid >> 5, lane = tid & 31;
  int r = blockIdx.x * 8 + w;
  if (r >= B_ * L_) return;
  int b = r / L_, l = r % L_;
  const float4* rp = (const float4*)(prompt + (size_t)r * D_);
  float ss = 0.f, dc[C_] = {0.f, 0.f, 0.f, 0.f, 0.f};
  #pragma unroll
  for (int g = 0; g < 4; ++g) {
    float4 v = rp[g * 32 + lane];
    ss += v.x * v.x + v.y * v.y + v.z * v.z + v.w * v.w;
    int base = (g * 32 + lane) * 4;
    #pragma unroll
    for (int c = 0; c < C_; ++c) {
      float4 f = *(const float4*)(fn + c * D_ + base);
      dc[c] += v.x * f.x + v.y * f.y + v.z * f.z + v.w * f.w;
    }
  }
  ss = wave_red(ss);
  #pragma unroll
  for (int c = 0; c < C_; ++c) dc[c] = wave_red(dc[c]);
  if (lane == 0) {
    float inv = 1.f / fmaxf(sqrtf(ss), 1e-12f);
    pinv[r] = inv;
    #pragma unroll
    for (int c = 0; c < C_; ++c) simc[((size_t)(b * C_ + c)) * L_ + l] = dc[c] * inv;
  }
}

// ---------------------------------------------------------------- K3: M[b,c,d] = sum_l simc*pinv*prompt
__global__ void k_M(const float* __restrict__ prompt, const float* __restrict__ simc,
                    const float* __restrict__ pinv, float* __restrict__ M) {
  int b = blockIdx.x / C_, c = blockIdx.x % C_, tid = threadIdx.x;
  __shared__ float coef[L_];
  if (tid < L_) coef[tid] = simc[((size_t)(b * C_ + c)) * L_ + tid] * pinv[b * L_ + tid];
  __syncthreads();
  for (int d = tid; d < D_; d += 256) {
    float a = 0.f;
    for (int l = 0; l < L_; ++l) a += prompt[((size_t)(b * L_ + l)) * D_ + d] * coef[l];
    M[(size_t)blockIdx.x * D_ + d] = a;
  }
}

// ---------------------------------------------------------------- K4: pass 1 over texture -> logits
__global__ void k_logits(const float* __restrict__ tex, const float* __restrict__ M, float* __restrict__ logits) {
  __shared__ __align__(16) float Mb[C_ * D_];
  int b = blockIdx.y, tid = threadIdx.x, w = tid >> 5, lane = tid & 31;
  for (int i = tid; i < C_ * D_; i += 256) Mb[i] = M[(size_t)b * (C_ * D_) + i];
  __syncthreads();
  int j = blockIdx.x * 8 + w;
  if (j >= J_) return;
  const float4* rp = (const float4*)(tex + ((size_t)b * J_ + j) * D_);
  float ss = 0.f, dc[C_] = {0.f, 0.f, 0.f, 0.f, 0.f};
  #pragma unroll
  for (int g = 0; g < 4; ++g) {
    float4 v = rp[g * 32 + lane];
    ss += v.x * v.x + v.y * v.y + v.z * v.z + v.w * v.w;
    int base = (g * 32 + lane) * 4;
    #pragma unroll
    for (int c = 0; c < C_; ++c) {
      float4 m = *(const float4*)(Mb + c * D_ + base);
      dc[c] += v.x * m.x + v.y * m.y + v.z * m.z + v.w * m.w;
    }
  }
  ss = wave_red(ss);
  #pragma unroll
  for (int c = 0; c < C_; ++c) dc[c] = wave_red(dc[c]);
  if (lane == 0) {
    float inv = 1.f / fmaxf(sqrtf(ss), 1e-12f);
    #pragma unroll
    for (int c = 0; c < C_; ++c) logits[((size_t)(b * C_ + c)) * J_ + j] = dc[c] * inv;
  }
}

// ---------------------------------------------------------------- K5: softmax stats over j per (b,c)
__global__ void k_smax(const float* __restrict__ logits, float* __restrict__ smax, float* __restrict__ ssum) {
  int bc = blockIdx.x, tid = threadIdx.x;
  __shared__ float red[256];
  const float* lg = logits + (size_t)bc * J_;
  float m = -3.4e38f;
  for (int i = tid; i < J_; i += 256) m = fmaxf(m, lg[i]);
  red[tid] = m; __syncthreads();
  for (int s = 128; s > 0; s >>= 1) { if (tid < s) red[tid] = fmaxf(red[tid], red[tid + s]); __syncthreads(); }
  float mm = red[0]; __syncthreads();
  float s = 0.f;
  for (int i = tid; i < J_; i += 256) s += expf(lg[i] - mm);
  red[tid] = s; __syncthreads();
  for (int st = 128; st > 0; st >>= 1) { if (tid < st) red[tid] += red[tid + st]; __syncthreads(); }
  if (tid == 0) { smax[bc] = mm; ssum[bc] = red[0]; }
}

// ---------------------------------------------------------------- zero helper
__global__ void k_zero(float* __restrict__ p, int n) {
  for (int i = blockIdx.x * blockDim.x + threadIdx.x; i < n; i += gridDim.x * blockDim.x) p[i] = 0.f;
}

// ---------------------------------------------------------------- K6: pass 2 over texture -> ftf (j-split, atomics)
#define JCHUNK 394
__global__ void k_ftf(const float* __restrict__ tex, const float* __restrict__ logits,
                      const float* __restrict__ smax, const float* __restrict__ ssum, float* __restrict__ ftf) {
  int b = blockIdx.x, dblk = blockIdx.y, jc = blockIdx.z, tid = threadIdx.x;
  int d = dblk * 128 + tid;
  int j0 = jc * JCHUNK, j1 = min(J_, j0 + JCHUNK);
  __shared__ float wl[C_][128];
  float acc[C_] = {0.f, 0.f, 0.f, 0.f, 0.f};
  float mx[C_];
  #pragma unroll
  for (int c = 0; c < C_; ++c) mx[c] = smax[b * C_ + c];
  for (int jj = j0; jj < j1; jj += 128) {
    int nj = min(128, j1 - jj);
    __syncthreads();
    if (tid < nj) {
      int j = jj + tid;
      #pragma unroll
      for (int c = 0; c < C_; ++c) wl[c][tid] = expf(logits[((size_t)(b * C_ + c)) * J_ + j] - mx[c]);
    }
    __syncthreads();
    const float* tp = tex + ((size_t)b * J_ + jj) * D_ + d;
    for (int u = 0; u < nj; ++u) {
      float t = tp[(size_t)u * D_];
      #pragma unroll
      for (int c = 0; c < C_; ++c) acc[c] += t * wl[c][u];
    }
  }
  #pragma unroll
  for (int c = 0; c < C_; ++c)
    atomicAdd(&ftf[((size_t)(b * C_ + c)) * D_ + d], acc[c] / ssum[b * C_ + c]);
}

// ---------------------------------------------------------------- K7: fq GEMM via V_WMMA_F32_16X16X4_F32
// fq[row,k] = sum_d (ftf[row,d]*eot[row/5,d]) * qw[k,d] + qb[k];  rows=160 (10 m-tiles), N=224 (14 n-tiles)
__global__ void k_fq(const float* __restrict__ ftf, const float* __restrict__ eot,
                     const float* __restrict__ qw, const float* __restrict__ qb, float* __restrict__ fq) {
  int wid = blockIdx.x * 4 + (threadIdx.x >> 5);
  int lane = threadIdx.x & 31;
  int tm = wid / 14, tn = wid % 14;
  int half = lane >> 4, m16 = lane & 15;
  int row = tm * 16 + m16;
  int bb = row / C_;
  int col = tn * 16 + m16;
  v8f acc = {0.f, 0.f, 0.f, 0.f, 0.f, 0.f, 0.f, 0.f};
  const float* fr = ftf + (size_t)row * D_;
  const float* er = eot + (size_t)bb * D_;
  const float* qr = qw + (size_t)col * D_;
  for (int k0 = 0; k0 < D_; k0 += 4) {
    int kb = k0 + 2 * half;
    v2f a; a.x = fr[kb] * er[kb]; a.y = fr[kb + 1] * er[kb + 1];
    v2f bf; bf.x = qr[kb]; bf.y = qr[kb + 1];
    acc = __builtin_amdgcn_wmma_f32_16x16x4_f32(false, a, false, bf, (short)0, acc, false, false);
  }
  #pragma unroll
  for (int r = 0; r < 8; ++r) {
    int mrow = tm * 16 + r + 8 * half;
    fq[(size_t)mrow * 224 + col] = acc[r] + qb[col];
  }
}

// ---------------------------------------------------------------- K8: cond_maps GEMM via WMMA (M padded 5->16)
__global__ void k_cmap(const float* __restrict__ fn, const float* __restrict__ cw,
                       const float* __restrict__ cb, float* __restrict__ cmaps) {
  int wid = blockIdx.x * 4 + (threadIdx.x >> 5);
  if (wid >= 343) return;  // 5488/16 tiles
  int lane = threadIdx.x & 31, half = lane >> 4, m16 = lane & 15;
  int n = wid * 16 + m16;
  v8f acc = {0.f, 0.f, 0.f, 0.f, 0.f, 0.f, 0.f, 0.f};
  const float* ar = fn + (size_t)m16 * D_;
  const float* br = cw + (size_t)n * D_;
  for (int k0 = 0; k0 < D_; k0 += 4) {
    int kb = k0 + 2 * half;
    v2f a;
    if (m16 < C_) { a.x = ar[kb]; a.y = ar[kb + 1]; } else { a.x = 0.f; a.y = 0.f; }
    v2f bf; bf.x = br[kb]; bf.y = br[kb + 1];
    acc = __builtin_amdgcn_wmma_f32_16x16x4_f32(false, a, false, bf, (short)0, acc, false, false);
  }
  #pragma unroll
  for (int r = 0; r < 8; ++r) {
    int m = r + 8 * half;
    if (m < C_) cmaps[(size_t)m * 5488 + n] = acc[r] + cb[n];
  }
}

// ---------------------------------------------------------------- K9: pooled + generated biases
__global__ void k_hb(const float* __restrict__ cmaps,
                     const float* w1, const float* b1, const float* w2, const float* b2,
                     const float* w3, const float* b3, const float* w4, const float* b4,
                     const float* w5, const float* b5, const float* w6, const float* b6,
                     float* f1b, float* f2b, float* f3b, float* f4b, float* f5w, float* f5b) {
  int c = blockIdx.x, tid = threadIdx.x;
  __shared__ float pl[112];
  if (tid < 112) {
    float s = 0.f;
    const float* p = cmaps + (size_t)c * 5488 + tid * 49;
    for (int i = 0; i < 49; ++i) s += p[i];
    pl[tid] = s * (1.f / 49.f);
  }
  __syncthreads();
  auto dot112 = [&](const float* wr) -> float {
    float a = 0.f;
    for (int i = 0; i < 112; ++i) a += pl[i] * wr[i];
    return a;
  };
  if (tid < 112)       { f1b[c * 112 + tid] = dot112(w1 + (size_t)tid * 112) + b1[tid]; }
  else if (tid < 168)  { int i = tid - 112; f2b[c * 56 + i] = dot112(w2 + (size_t)i * 112) + b2[i]; }
  else if (tid < 196)  { int i = tid - 168; f3b[c * 28 + i] = dot112(w3 + (size_t)i * 112) + b3[i]; }
  else if (tid < 210)  { int i = tid - 196; f4b[c * 14 + i] = dot112(w4 + (size_t)i * 112) + b4[i]; }
  else if (tid < 224)  { int i = tid - 210; f5w[c * 14 + i] = dot112(w5 + (size_t)i * 112) + b5[i]; }
  else if (tid == 224) { f5b[c] = dot112(w6) + b6[0]; }
}

// ---------------------------------------------------------------- K10: 3x3 SAME conv on 7x7 maps (hypernet weights)
__global__ void k_conv(const float* __restrict__ cmaps, const float* __restrict__ w,
                       const float* __restrict__ bias, float* __restrict__ out, int O) {
  __shared__ float mp[112 * 49];
  int c = blockIdx.y;
  int tid = threadIdx.y * 49 + threadIdx.x;
  for (int i = tid; i < 5488; i += 392) mp[i] = cmaps[(size_t)c * 5488 + i];
  __syncthreads();
  int o = blockIdx.x * 8 + threadIdx.y;
  if (o >= O) return;
  int px = threadIdx.x % 7, py = threadIdx.x / 7;
  float a = bias[o];
  const float* wr = w + (size_t)o * 112 * 9;
  for (int i = 0; i < 112; ++i) {
    const float* mrow = mp + i * 49;
    const float* wi = wr + i * 9;
    #pragma unroll
    for (int ky = 0; ky < 3; ++ky) {
      int yy = py + ky - 1;
      if (yy < 0 || yy > 6) continue;
      #pragma unroll
      for (int kx = 0; kx < 3; ++kx) {
        int xx = px + kx - 1;
        if (xx < 0 || xx > 6) continue;
        a += mrow[yy * 7 + xx] * wi[ky * 3 + kx];
      }
    }
  }
  out[(size_t)c * O * 49 + (size_t)o * 49 + threadIdx.x] = a;
}

// ---------------------------------------------------------------- K11: TargetNet per (b,c)
__global__ void k_target(const float* __restrict__ fq,
                         const float* f1w, const float* f1b, const float* f2w, const float* f2b,
                         const float* f3w, const float* f3b, const float* f4w, const float* f4b,
                         const float* f5w, const float* f5b, float* __restrict__ out) {
  int c = blockIdx.x % C_, tid = threadIdx.x;
  __shared__ float fv[224], h1[112], h2[56], h3[28], h4[14];
  if (tid < 224) fv[tid] = fq[(size_t)blockIdx.x * 224 + tid];
  __syncthreads();
  if (tid < 112) {
    float a = f1b[c * 112 + tid];
    const float* wr = f1w + (size_t)c * 25088 + (size_t)tid * 224;
    for (int k = 0; k < 224; ++k) a += fv[k] * wr[k];
    h1[tid] = sigmoidf(a);
  }
  __syncthreads();
  if (tid < 56) {
    float a = f2b[c * 56 + tid];
    const float* wr = f2w + (size_t)c * 6272 + (size_t)tid * 112;
    for (int k = 0; k < 112; ++k) a += h1[k] * wr[k];
    h2[tid] = sigmoidf(a);
  }
  __syncthreads();
  if (tid < 28) {
    float a = f3b[c * 28 + tid];
    const float* wr = f3w + (size_t)c * 1568 + (size_t)tid * 56;
    for (int k = 0; k < 56; ++k) a += h2[k] * wr[k];
    h3[tid] = sigmoidf(a);
  }
  __syncthreads();
  if (tid < 14) {
    float a = f4b[c * 14 + tid];
    const float* wr = f4w + (size_t)c * 392 + (size_t)tid * 28;
    for (int k = 0; k < 28; ++k) a += h3[k] * wr[k];
    h4[tid] = sigmoidf(a);
  }
  __syncthreads();
  if (tid == 0) {
    float a = f5b[c];
    for (int p = 0; p < 14; ++p) a += h4[p] * f5w[c * 14 + p];
    out[blockIdx.x] = a;
  }
}

// ----------------------------------------------------------------
extern "C" void kernel_launch(void* const* d_in, const int* in_sizes, int n_in,
                              void* d_out, int out_size, void* d_ws, size_t ws_size,
                              hipStream_t stream) {
  const float* tex   = (const float*)d_in[0];   // [32,1182,512]
  const float* prom  = (const float*)d_in[1];   // [32,77,512]
  const float* eot   = (const float*)d_in[2];   // [32,512]
  const float* cond  = (const float*)d_in[3];   // [5,512]
  const float* qw    = (const float*)d_in[4];   // [224,512]
  const float* qb    = (const float*)d_in[5];   // [224]
  const float* ccw   = (const float*)d_in[6];   // [5488,512]
  const float* ccb   = (const float*)d_in[7];   // [5488]
  const float* c1w   = (const float*)d_in[8];   const float* c1b = (const float*)d_in[9];
  const float* c2w   = (const float*)d_in[10];  const float* c2b = (const float*)d_in[11];
  const float* c3w   = (const float*)d_in[12];  const float* c3b = (const float*)d_in[13];
  const float* c4w   = (const float*)d_in[14];  const float* c4b = (const float*)d_in[15];
  const float* b1w   = (const float*)d_in[16];  const float* b1b = (const float*)d_in[17];
  const float* b2w   = (const float*)d_in[18];  const float* b2b = (const float*)d_in[19];
  const float* b3w   = (const float*)d_in[20];  const float* b3b = (const float*)d_in[21];
  const float* b4w   = (const float*)d_in[22];  const float* b4b = (const float*)d_in[23];
  const float* w5w   = (const float*)d_in[24];  const float* w5b = (const float*)d_in[25];
  const float* b5w   = (const float*)d_in[26];  const float* b5b = (const float*)d_in[27];
  float* out = (float*)d_out;

  float* w = (float*)d_ws;
  size_t o = 0;
  float* fn     = w + o; o += C_ * D_;          // 2560
  float* pinv   = w + o; o += B_ * L_;          // 2464
  float* simc   = w + o; o += B_ * C_ * L_;     // 12320
  float* M      = w + o; o += B_ * C_ * D_;     // 81920
  float* logits = w + o; o += (size_t)B_ * C_ * J_; // 189120
  float* smax   = w + o; o += B_ * C_;
  float* ssum   = w + o; o += B_ * C_;
  float* ftf    = w + o; o += B_ * C_ * D_;     // 81920
  float* fq     = w + o; o += B_ * C_ * 224;    // 35840
  float* cmaps  = w + o; o += C_ * 5488;        // 27440
  float* f1b    = w + o; o += C_ * 112;
  float* f2b    = w + o; o += C_ * 56;
  float* f3b    = w + o; o += C_ * 28;
  float* f4b    = w + o; o += C_ * 14;
  float* f5w    = w + o; o += C_ * 14;
  float* f5b    = w + o; o += 8;
  float* f1w    = w + o; o += C_ * 25088;       // 125440
  float* f2w    = w + o; o += C_ * 6272;
  float* f3w    = w + o; o += C_ * 1568;
  float* f4w    = w + o; o += C_ * 392;

  // condition path
  k_cond<<<1, 320, 0, stream>>>(cond, fn, out + 160);
  k_cmap<<<86, 128, 0, stream>>>(fn, ccw, ccb, cmaps);                          // WMMA
  k_hb<<<C_, 256, 0, stream>>>(cmaps, b1w, b1b, b2w, b2b, b3w, b3b, b4w, b4b,
                               w5w, w5b, b5w, b5b, f1b, f2b, f3b, f4b, f5w, f5b);
  k_conv<<<dim3(64, C_), dim3(49, 8), 0, stream>>>(cmaps, c1w, c1b, f1w, 512);
  k_conv<<<dim3(16, C_), dim3(49, 8), 0, stream>>>(cmaps, c2w, c2b, f2w, 128);
  k_conv<<<dim3(4, C_),  dim3(49, 8), 0, stream>>>(cmaps, c3w, c3b, f3w, 32);
  k_conv<<<dim3(1, C_),  dim3(49, 8), 0, stream>>>(cmaps, c4w, c4b, f4w, 8);

  // attention / texture path
  k_prompt<<<(B_ * L_ + 7) / 8, 256, 0, stream>>>(prom, fn, pinv, simc);
  k_M<<<B_ * C_, 256, 0, stream>>>(prom, simc, pinv, M);
  k_logits<<<dim3((J_ + 7) / 8, B_), 256, 0, stream>>>(tex, M, logits);
  k_smax<<<B_ * C_, 256, 0, stream>>>(logits, smax, ssum);
  k_zero<<<80, 256, 0, stream>>>(ftf, B_ * C_ * D_);
  k_ftf<<<dim3(B_, 4, 3), 128, 0, stream>>>(tex, logits, smax, ssum, ftf);
  k_fq<<<35, 128, 0, stream>>>(ftf, eot, qw, qb, fq);                           // WMMA

  // target net
  k_target<<<B_ * C_, 256, 0, stream>>>(fq, f1w, f1b, f2w, f2b, f3w, f3b,
                                        f4w, f4b, f5w, f5b, out);
}